// ParallelMetaAttention_13374528160071
// MI455X (gfx1250) — compile-verified
//
#include <hip/hip_runtime.h>
#include <math.h>

// ---------------- problem constants ----------------
#define BATCH 2
#define DIMC  64
#define IMG_H 512
#define IMG_W 512
#define HEADS 8
#define NKEYS 4
#define MDIM  4
#define CHN   8
#define HW    (IMG_H * IMG_W)
#define EPSN  1e-12f

// ---------------- tiling ----------------
#define TW   32            // tile width  (inner)
#define TH   8             // tile height (inner)
#define HLW  (TW + 2)      // 34 halo width
#define HLH  (TH + 2)      // 10 halo height
#define HLP  (HLW * HLH)   // 340 halo pixels
#define NTILE ((HLP + 15) / 16)   // 22 WMMA N-tiles over halo pixels
#define HLPAD (NTILE * 16)        // 352 padded pixel stride (kv1 buffer)
#define XPR  712           // pair-row stride (floats) for interleaved x staging
                           //   = 2*HLPAD padded to 8 mod 64 banks, keeps 8B align
#define GPR  520           // pair-row stride (floats) for interleaved gelu buffer
#define ZCHUNK 1024        // pixels per k_zsum block (4 per thread -> b128 loads)

typedef __attribute__((ext_vector_type(2))) float v2f;
typedef __attribute__((ext_vector_type(8))) float v8f;

// f32 WMMA: D(16x16) = A(16x4) * B(4x16) + C
// A: lane<16 holds {K0,K1} of row M=lane; lane>=16 holds {K2,K3} of row M=lane-16
// B: lane<16 holds rows K0,K1 at col N=lane; lane>=16 holds rows K2,K3 at col N=lane-16
// C/D: vgpr r -> (M=r, N=lane) for lane<16, (M=r+8, N=lane-16) for lane>=16
__device__ __forceinline__ v8f wmma_f32_16x16x4(v2f a, v2f b, v8f c) {
  return __builtin_amdgcn_wmma_f32_16x16x4_f32(false, a, false, b, (short)0, c,
                                               false, false);
}

// ---------------- kernel 0: zero global accumulators ----------------
__global__ void k_init(float* __restrict__ normsq, float* __restrict__ Z) {
  int i = threadIdx.x;
  if (i < BATCH * DIMC) normsq[i] = 0.f;
  if (i < BATCH * NKEYS * HEADS) Z[i] = 0.f;
}

// ---------------- kernel 1: fused 1x1 conv (WMMA) + 3x3 depthwise + k-norm^2 ----------------
__global__ __launch_bounds__(256) void k_kvconv(
    const float* __restrict__ x, const float* __restrict__ w1,
    const float* __restrict__ b1, const float* __restrict__ dww,
    const float* __restrict__ dwb, float* __restrict__ kv,
    float* __restrict__ normsq) {
  // x tile staged as interleaved channel PAIRS so each lane's WMMA B fragment
  // (channels kb, kb+1 at one pixel) is a single ds_load_b64.
  __shared__ float xs[(DIMC / 2) * XPR];  // 89 KB
  __shared__ float kv1[16 * HLPAD];       // 22 KB: 1x1 result, one 16-ch block
  __shared__ float ksq[16];

  const int tid = threadIdx.x;
  const int lane = tid & 31;
  const int wid = tid >> 5;
  const int bz = blockIdx.z;
  const int x0 = blockIdx.x * TW, y0 = blockIdx.y * TH;

  // stage x halo tile (zero-padded at image borders and LDS pad region)
  for (int idx = tid; idx < DIMC * HLPAD; idx += 256) {
    int ic = idx / HLPAD, lp = idx - ic * HLPAD;
    float v = 0.f;
    if (lp < HLP) {
      int ly = lp / HLW, lx = lp - ly * HLW;
      int gy = y0 + ly - 1, gx = x0 + lx - 1;
      if (gy >= 0 && gy < IMG_H && gx >= 0 && gx < IMG_W)
        v = x[(size_t)(bz * DIMC + ic) * HW + (size_t)gy * IMG_W + gx];
    }
    xs[(ic >> 1) * XPR + lp * 2 + (ic & 1)] = v;
  }
  __syncthreads();

  const int n = lane & 15;
  const int khpair = (lane < 16) ? 0 : 1;  // which K pair of the 4-wide K step
  const int mofs   = (lane < 16) ? 0 : 8;
  const int py = tid >> 5, px = tid & 31;  // this thread's inner pixel

  for (int ob = 0; ob < 8; ++ob) {  // 8 blocks of 16 output channels
    // --- 1x1 conv for 16 channels over all halo pixels, via f32 WMMA ---
    for (int nt = wid; nt < NTILE; nt += 8) {  // wave-uniform: EXEC stays full
      int pb = nt * 16;
      v8f acc = {0.f, 0.f, 0.f, 0.f, 0.f, 0.f, 0.f, 0.f};
      for (int k4 = 0; k4 < 16; ++k4) {
        int kb = k4 * 4 + khpair * 2;
        int m = ob * 16 + n;
        v2f a;
        a.x = w1[m * DIMC + kb];
        a.y = w1[m * DIMC + kb + 1];
        v2f bf = *(const v2f*)&xs[(k4 * 2 + khpair) * XPR + (pb + n) * 2];
        acc = wmma_f32_16x16x4(a, bf, acc);
      }
      for (int r = 0; r < 8; ++r) {
        int m = r + mofs;
        kv1[m * HLPAD + pb + n] = acc[r] + b1[ob * 16 + m];
      }
    }
    if (tid < 16) ksq[tid] = 0.f;
    __syncthreads();

    // --- 3x3 depthwise over inner 32x8 pixels; accumulate k^2 for L2 norm ---
    for (int c = 0; c < 16; ++c) {
      int oc = ob * 16 + c;
      const float* wd = dww + oc * 9;
      float s = dwb[oc];
#pragma unroll
      for (int dy = 0; dy < 3; ++dy)
#pragma unroll
        for (int dx = 0; dx < 3; ++dx)
          s += wd[dy * 3 + dx] * kv1[c * HLPAD + (py + dy) * HLW + (px + dx)];
      kv[(size_t)(bz * 2 * DIMC + oc) * HW + (size_t)(y0 + py) * IMG_W + (x0 + px)] = s;
      float s2 = s * s;
      for (int off = 16; off > 0; off >>= 1) s2 += __shfl_xor(s2, off, 32);
      if (lane == 0) atomicAdd(&ksq[c], s2);
    }
    __syncthreads();
    if (ob < 4 && tid < 16)  // only the k half (first 64 channels) is normalized
      atomicAdd(&normsq[bz * DIMC + ob * 16 + tid], ksq[tid]);
    __syncthreads();
  }
}

// ---------------- kernel 2: fold q-proj, sign-normalize, 1/||k||, temps ----------------
__global__ void k_prep(const float* __restrict__ meta, const float* __restrict__ Wq,
                       const float* __restrict__ bq, const float* __restrict__ temps,
                       const float* __restrict__ normsq, float* __restrict__ wcoef) {
  int idx = blockIdx.x * blockDim.x + threadIdx.x;
  if (idx >= BATCH * NKEYS * HEADS * CHN) return;
  int c = idx & 7, e = (idx >> 3) & 7, nk = (idx >> 6) & 3, b = idx >> 8;
  int d = e * CHN + c;  // flat dim index of q/k channel
  float q = bq[nk * DIMC + d];
  for (int m = 0; m < MDIM; ++m)
    q += meta[(b * NKEYS + nk) * MDIM + m] * Wq[(nk * DIMC + d) * MDIM + m];
  float qn = q / fmaxf(fabsf(q), EPSN);        // F.normalize over singleton axis
  float nrm = sqrtf(normsq[b * DIMC + d]);     // L2 norm of k row over all pixels
  wcoef[idx] = temps[nk * HEADS + e] * qn / fmaxf(nrm, EPSN);
}

// ---------------- kernel 3: softmax denominators Z[b,n,e] = sum_s exp(t) ----------------
// |t| <= temp * sum_c |q_hat|*|k_hat| <= 8*temp since k rows are unit-norm -> no max-shift.
// Each thread owns 4 consecutive pixels so every channel read is one b128 load.
__global__ __launch_bounds__(256) void k_zsum(const float* __restrict__ kv,
                                              const float* __restrict__ wcoef,
                                              float* __restrict__ Z) {
  __shared__ float wc[NKEYS * HEADS * CHN];
  __shared__ float zred[8][NKEYS * HEADS];
  const int tid = threadIdx.x, lane = tid & 31, wid = tid >> 5;
  const int b = blockIdx.y;
  const int s0 = blockIdx.x * ZCHUNK + tid * 4;
  if (tid < NKEYS * HEADS * CHN) wc[tid] = wcoef[b * NKEYS * HEADS * CHN + tid];
  __syncthreads();

  float zl[NKEYS * HEADS];
#pragma unroll
  for (int i = 0; i < NKEYS * HEADS; ++i) zl[i] = 0.f;
  const float* kb = kv + (size_t)b * 2 * DIMC * HW;

#pragma unroll
  for (int e = 0; e < HEADS; ++e) {
    float4 kc[CHN];
#pragma unroll
    for (int c = 0; c < CHN; ++c)
      kc[c] = *(const float4*)&kb[(size_t)(e * CHN + c) * HW + s0];
#pragma unroll
    for (int nk = 0; nk < NKEYS; ++nk) {
      float tx = 0.f, ty = 0.f, tz = 0.f, tw = 0.f;
#pragma unroll
      for (int c = 0; c < CHN; ++c) {
        float w = wc[(nk * HEADS + e) * CHN + c];
        tx += w * kc[c].x;
        ty += w * kc[c].y;
        tz += w * kc[c].z;
        tw += w * kc[c].w;
      }
      zl[nk * HEADS + e] += __expf(tx) + __expf(ty) + __expf(tz) + __expf(tw);
    }
  }
#pragma unroll
  for (int i = 0; i < NKEYS * HEADS; ++i) {
    float v = zl[i];
    for (int off = 16; off > 0; off >>= 1) v += __shfl_xor(v, off, 32);
    if (lane == 0) zred[wid][i] = v;
  }
  __syncthreads();
  if (tid < NKEYS * HEADS) {
    float v = 0.f;
    for (int w = 0; w < 8; ++w) v += zred[w][tid];
    atomicAdd(&Z[b * NKEYS * HEADS + tid], v);
  }
}

// ---------------- kernel 4: fused softmax-apply + grouped 3x3 + GELU + 1x1 (WMMA) ----------------
__global__ __launch_bounds__(256) void k_attn_cc(
    const float* __restrict__ kv, const float* __restrict__ wcoef,
    const float* __restrict__ Z,
    const float* __restrict__ c1w, const float* __restrict__ c1b,
    const float* __restrict__ c2w, const float* __restrict__ c2b,
    float* __restrict__ out) {
  __shared__ float o8[CHN * HLPAD];        // 11 KB: attn-weighted values, 8 ch x halo px
  __shared__ float g[(DIMC / 2) * GPR];    // 65 KB: GELU result, channel-PAIR interleaved
  __shared__ float wc[NKEYS * HEADS * CHN];
  __shared__ float invZ[NKEYS * HEADS];

  const int tid = threadIdx.x, lane = tid & 31, wid = tid >> 5;
  const int bz = blockIdx.z;
  const int x0 = blockIdx.x * TW, y0 = blockIdx.y * TH;
  if (tid < NKEYS * HEADS * CHN) wc[tid] = wcoef[bz * NKEYS * HEADS * CHN + tid];
  if (tid < NKEYS * HEADS) invZ[tid] = 1.f / Z[bz * NKEYS * HEADS + tid];
  __syncthreads();

  const float* kb = kv + (size_t)bz * 2 * DIMC * HW;

  // phase 1: out8[c,s] = sum_{n,e} softmax(t)[n,e,s] * v[e,c,s]   (halo pixels)
  for (int lp = tid; lp < HLP; lp += 256) {
    int ly = lp / HLW, lx = lp - ly * HLW;
    int gy = y0 + ly - 1, gx = x0 + lx - 1;
    float acc8[CHN];
#pragma unroll
    for (int c = 0; c < CHN; ++c) acc8[c] = 0.f;
    if (gy >= 0 && gy < IMG_H && gx >= 0 && gx < IMG_W) {
      size_t s = (size_t)gy * IMG_W + gx;
#pragma unroll
      for (int e = 0; e < HEADS; ++e) {
        float kc[CHN];
#pragma unroll
        for (int c = 0; c < CHN; ++c) kc[c] = kb[(size_t)(e * CHN + c) * HW + s];
        float pe = 0.f;
#pragma unroll
        for (int nk = 0; nk < NKEYS; ++nk) {
          float t = 0.f;
#pragma unroll
          for (int c = 0; c < CHN; ++c) t += wc[(nk * HEADS + e) * CHN + c] * kc[c];
          pe += __expf(t) * invZ[nk * HEADS + e];
        }
#pragma unroll
        for (int c = 0; c < CHN; ++c)
          acc8[c] += pe * kb[(size_t)(DIMC + e * CHN + c) * HW + s];
      }
    }
#pragma unroll
    for (int c = 0; c < CHN; ++c) o8[c * HLPAD + lp] = acc8[c];
  }
  __syncthreads();

  // phase 2: grouped 3x3 conv (8 -> 64, groups=8) + exact GELU, pair-interleaved store
  {
    const int py = tid >> 5, px = tid & 31;
    for (int j = 0; j < DIMC; ++j) {
      int cg = j >> 3;
      const float* wj = c1w + j * 9;
      float s = c1b[j];
#pragma unroll
      for (int dy = 0; dy < 3; ++dy)
#pragma unroll
        for (int dx = 0; dx < 3; ++dx)
          s += wj[dy * 3 + dx] * o8[cg * HLPAD + (py + dy) * HLW + (px + dx)];
      g[(j >> 1) * GPR + tid * 2 + (j & 1)] =
          0.5f * s * (1.f + erff(s * 0.70710678118654752f));
    }
  }
  __syncthreads();

  // phase 3: 1x1 conv 64 -> 64 via f32 WMMA (B fragment = one ds_load_b64)
  {
    const int n = lane & 15;
    const int khpair = (lane < 16) ? 0 : 1;
    const int mofs   = (lane < 16) ? 0 : 8;
    for (int t = wid; t < 64; t += 8) {  // 4 M-tiles x 16 N-tiles, wave-uniform
      int mt = t >> 4, nt = t & 15;
      int pb = nt * 16;
      v8f acc = {0.f, 0.f, 0.f, 0.f, 0.f, 0.f, 0.f, 0.f};
      for (int k4 = 0; k4 < 16; ++k4) {
        int kbn = k4 * 4 + khpair * 2;
        int m = mt * 16 + n;
        v2f a;
        a.x = c2w[m * DIMC + kbn];
        a.y = c2w[m * DIMC + kbn + 1];
        v2f bf = *(const v2f*)&g[(k4 * 2 + khpair) * GPR + (pb + n) * 2];
        acc = wmma_f32_16x16x4(a, bf, acc);
      }
      int p = pb + n;
      int py = p >> 5, px = p & 31;
      size_t so = (size_t)(y0 + py) * IMG_W + (x0 + px);
#pragma unroll
      for (int r = 0; r < 8; ++r) {
        int m = mt * 16 + r + mofs;
        out[(size_t)(bz * DIMC + m) * HW + so] = acc[r] + c2b[m];
      }
    }
  }
}

// ---------------- host launcher ----------------
extern "C" void kernel_launch(void* const* d_in, const int* in_sizes, int n_in,
                              void* d_out, int out_size, void* d_ws, size_t ws_size,
                              hipStream_t stream) {
  const float* x     = (const float*)d_in[0];
  const float* meta  = (const float*)d_in[1];
  const float* Wq    = (const float*)d_in[2];
  const float* bq    = (const float*)d_in[3];
  const float* temps = (const float*)d_in[4];
  const float* kvw   = (const float*)d_in[5];
  const float* kvb   = (const float*)d_in[6];
  const float* dww   = (const float*)d_in[7];
  const float* dwb   = (const float*)d_in[8];
  const float* c1w   = (const float*)d_in[9];
  const float* c1b   = (const float*)d_in[10];
  const float* c2w   = (const float*)d_in[11];
  const float* c2b   = (const float*)d_in[12];
  float* outp = (float*)d_out;

  // workspace layout (floats): kv[2*128*HW] | normsq[128] | wcoef[512] | Z[64]
  float* kv     = (float*)d_ws;
  float* normsq = kv + (size_t)BATCH * 2 * DIMC * HW;
  float* wcoef  = normsq + BATCH * DIMC;
  float* Z      = wcoef + BATCH * NKEYS * HEADS * CHN;

  hipLaunchKernelGGL(k_init, dim3(1), dim3(128), 0, stream, normsq, Z);
  hipLaunchKernelGGL(k_kvconv, dim3(IMG_W / TW, IMG_H / TH, BATCH), dim3(256), 0,
                     stream, x, kvw, kvb, dww, dwb, kv, normsq);
  hipLaunchKernelGGL(k_prep, dim3(2), dim3(256), 0, stream, meta, Wq, bq, temps,
                     normsq, wcoef);
  hipLaunchKernelGGL(k_zsum, dim3(HW / ZCHUNK, BATCH), dim3(256), 0, stream, kv,
                     wcoef, Z);
  hipLaunchKernelGGL(k_attn_cc, dim3(IMG_W / TW, IMG_H / TH, BATCH), dim3(256), 0,
                     stream, kv, wcoef, Z, c1w, c1b, c2w, c2b, outp);
}